// PathConv_51041391346227
// MI455X (gfx1250) — compile-verified
//
#include <hip/hip_runtime.h>
#include <hip/hip_bf16.h>

// ---------------------------------------------------------------------------
// PathConv (GAT-style pull + dual MLP) for MI455X / gfx1250, wave32.
// Only rows in `targets` are ever read from the output, so:
//   - compact target nodes to dense ids (needed ~<= T of N nodes)
//   - process only edges whose dst is a target (~18% of E)
//   - run the 64->256->64 MLPs only on T rows, via V_WMMA_F32_16X16X4_F32
//     (fp32-exact matrix path on CDNA5).
// ---------------------------------------------------------------------------

typedef float v2f __attribute__((ext_vector_type(2)));
typedef float v8f __attribute__((ext_vector_type(8)));

#define WMMA_F32_16x16x4(a, b, c) \
    __builtin_amdgcn_wmma_f32_16x16x4_f32(false, (a), false, (b), (short)0, (c), false, false)

#ifndef PC_D
#define PC_D   64
#define PC_HID 256
#endif

#define HSTRIDE 250   // LDS row stride (floats): even (8B-align), 250%64=58, 8*250%64=16 -> conflict-free

// ---------------------------------------------------------------------------
// Phase 0: init / mark / compact
// ---------------------------------------------------------------------------
__global__ void k_init(int* __restrict__ mark, unsigned* __restrict__ counter, int N) {
    int i = blockIdx.x * blockDim.x + threadIdx.x;
    if (i < N) mark[i] = 0;
    if (i == 0) *counter = 0u;
}

__global__ void k_mark(const int* __restrict__ targets, int* __restrict__ mark, int T) {
    int i = blockIdx.x * blockDim.x + threadIdx.x;
    if (i < T) mark[targets[i]] = 1;
}

__global__ void k_assign(const int* __restrict__ mark, int* __restrict__ map,
                         unsigned* __restrict__ counter,
                         float* __restrict__ m_c, float* __restrict__ den_c,
                         float* __restrict__ hn, int N) {
    int n = blockIdx.x * blockDim.x + threadIdx.x;
    if (n >= N) return;
    if (mark[n]) {
        int id = (int)atomicAdd(counter, 1u);
        map[n] = id;
        m_c[id]  = 0.0f;   // e >= 0 after relu, so 0 is the correct identity/empty value
        den_c[id] = 0.0f;
        float4* row = (float4*)(hn + (size_t)id * PC_D);
#pragma unroll
        for (int j = 0; j < PC_D / 4; ++j) row[j] = make_float4(0.f, 0.f, 0.f, 0.f);
    } else {
        map[n] = -1;
    }
}

// ---------------------------------------------------------------------------
// Phase 1: per-node attention score  s[n] = relu(h[n] . attn)
// ---------------------------------------------------------------------------
__global__ void k_score(const float* __restrict__ h, const float* __restrict__ attn,
                        float* __restrict__ s, int N) {
    int n = blockIdx.x * blockDim.x + threadIdx.x;
    if (n >= N) return;
    const float4* hp = (const float4*)(h + (size_t)n * PC_D);
    const float4* ap = (const float4*)attn;
    float acc = 0.f;
#pragma unroll
    for (int k = 0; k < PC_D / 4; ++k) {
        float4 a = ap[k];
        float4 v = hp[k];
        acc += a.x * v.x + a.y * v.y + a.z * v.z + a.w * v.w;
    }
    s[n] = fmaxf(acc, 0.f);
}

// ---------------------------------------------------------------------------
// Phase 2: segment softmax over dst (target nodes only)
// ---------------------------------------------------------------------------
__global__ void k_max(const int* __restrict__ src, const int* __restrict__ dst,
                      const int* __restrict__ map, const float* __restrict__ s,
                      float* __restrict__ m_c, int E) {
    int e = blockIdx.x * blockDim.x + threadIdx.x;
    if (e >= E) return;
    int mp = map[dst[e]];
    if (mp < 0) return;
    // s >= 0, so uint-bit compare == float compare
    atomicMax((unsigned*)(m_c + mp), __float_as_uint(s[src[e]]));
}

__global__ void k_denom(const int* __restrict__ src, const int* __restrict__ dst,
                        const int* __restrict__ map, const float* __restrict__ s,
                        const float* __restrict__ m_c, float* __restrict__ den_c, int E) {
    int e = blockIdx.x * blockDim.x + threadIdx.x;
    if (e >= E) return;
    int mp = map[dst[e]];
    if (mp < 0) return;
    float ex = expf(s[src[e]] - m_c[mp]);
    atomicAdd(den_c + mp, ex);
}

__global__ void k_coef(const int* __restrict__ src, const int* __restrict__ dst,
                       const int* __restrict__ map, const float* __restrict__ s,
                       const float* __restrict__ m_c, const float* __restrict__ den_c,
                       float* __restrict__ ac, int E) {
    int e = blockIdx.x * blockDim.x + threadIdx.x;
    if (e >= E) return;
    int mp = map[dst[e]];
    float a = 0.f;                 // 0 marks "edge not needed" (valid a is strictly > 0)
    if (mp >= 0) a = expf(s[src[e]] - m_c[mp]) / den_c[mp];
    ac[e] = a;
}

// ---------------------------------------------------------------------------
// Phase 3: weighted scatter  hn[map[dst]] += a * h[src]   (16 threads/edge)
// ---------------------------------------------------------------------------
__global__ void k_scatter(const int* __restrict__ src, const int* __restrict__ dst,
                          const int* __restrict__ map, const float* __restrict__ ac,
                          const float* __restrict__ h, float* __restrict__ hn, int E) {
    long long i = (long long)blockIdx.x * blockDim.x + threadIdx.x;
    long long tot = (long long)E * 16;
    if (i >= tot) return;
    int e = (int)(i >> 4);
    float a = ac[e];
    if (a <= 0.f) return;
    int k4 = ((int)i & 15) * 4;
    int sn = src[e];
    int mp = map[dst[e]];
    const float4 v = *(const float4*)(h + (size_t)sn * PC_D + k4);
    float* b = hn + (size_t)mp * PC_D + k4;
    atomicAdd(b + 0, a * v.x);
    atomicAdd(b + 1, a * v.y);
    atomicAdd(b + 2, a * v.z);
    atomicAdd(b + 3, a * v.w);
}

// ---------------------------------------------------------------------------
// Phase 4: fused dual MLP + relu on T target rows, fp32 WMMA.
// Block = 128 threads = 4 waves; each wave owns 16 output rows.
// ---------------------------------------------------------------------------
__launch_bounds__(128)
__global__ void k_mlp(const float* __restrict__ hn,   const float* __restrict__ cell,
                      const float* __restrict__ Wn1,  const float* __restrict__ bn1,
                      const float* __restrict__ Wn2,  const float* __restrict__ bn2,
                      const float* __restrict__ Ws1,  const float* __restrict__ bs1,
                      const float* __restrict__ Ws2,  const float* __restrict__ bs2,
                      const int*   __restrict__ targets, const int* __restrict__ map,
                      float* __restrict__ out, int T) {
    __shared__ float Hbuf[4 * 16 * HSTRIDE];   // per-wave 16 x HID hidden activations

    const int lane = threadIdx.x & 31;
    const int wave = threadIdx.x >> 5;
    const int col  = lane & 15;            // N index in tile == M row for A-fragments
    const int khi  = (lane >> 4) << 1;     // 0 or 2: which K-pair of the 4-chunk this lane holds
    const int rbase = blockIdx.x * 64 + wave * 16;

    // Row this lane sources A-fragments for (clamped for the tail; stores are guarded).
    const int myr = rbase + col;
    const int ic  = myr < (T - 1) ? myr : (T - 1);
    const int t   = targets[ic];
    const int mp  = map[t];                // always >= 0 for target nodes
    const float* xn = hn   + (size_t)mp * PC_D;
    const float* xs = cell + (size_t)t  * PC_D;

    float* Hl = &Hbuf[wave * 16 * HSTRIDE];

    v8f acc2[4];
#pragma unroll
    for (int j = 0; j < 4; ++j) acc2[j] = (v8f){0.f, 0.f, 0.f, 0.f, 0.f, 0.f, 0.f, 0.f};

    for (int pass = 0; pass < 2; ++pass) {
        const float* X  = pass ? xs  : xn;
        const float* W1 = pass ? Ws1 : Wn1;
        const float* B1 = pass ? bs1 : bn1;
        const float* W2 = pass ? Ws2 : Wn2;

        // Preload this lane's half of the input row: K-pairs for all 16 k-chunks.
        v2f xa[16];
#pragma unroll
        for (int kk = 0; kk < 16; ++kk) {
            int k0 = kk * 4 + khi;
            xa[kk].x = X[k0];
            xa[kk].y = X[k0 + 1];
        }

        __syncthreads();   // protect Hbuf reuse between passes

        // ---- Layer 1: H = relu(X @ W1 + b1), 16x256 per wave ----
        for (int nt = 0; nt < 16; ++nt) {
            const int nn = nt * 16;
            v8f c = (v8f){0.f, 0.f, 0.f, 0.f, 0.f, 0.f, 0.f, 0.f};
#pragma unroll
            for (int kk = 0; kk < 16; ++kk) {
                int k0 = kk * 4 + khi;
                v2f b;
                b.x = W1[(size_t)k0 * PC_HID + nn + col];
                b.y = W1[(size_t)(k0 + 1) * PC_HID + nn + col];
                c = WMMA_F32_16x16x4(xa[kk], b, c);
            }
            const float bias  = B1[nn + col];
            const int   mbase = (lane < 16) ? 0 : 8;
#pragma unroll
            for (int j = 0; j < 8; ++j) {
                Hl[(mbase + j) * HSTRIDE + nn + col] = fmaxf(c[j] + bias, 0.f);
            }
        }

        __syncthreads();

        // ---- Layer 2: O += H @ W2, 16x64 per wave, accumulated over both MLPs ----
        for (int nt = 0; nt < 4; ++nt) {
            const int nn = nt * 16;
            v8f c = acc2[nt];
#pragma unroll 8
            for (int kk = 0; kk < 64; ++kk) {
                int k0 = kk * 4 + khi;
                v2f a, b;
                a.x = Hl[col * HSTRIDE + k0];
                a.y = Hl[col * HSTRIDE + k0 + 1];
                b.x = W2[(size_t)k0 * PC_D + nn + col];
                b.y = W2[(size_t)(k0 + 1) * PC_D + nn + col];
                c = WMMA_F32_16x16x4(a, b, c);
            }
            acc2[nt] = c;
        }
    }

    // ---- Epilogue: out = relu(o_n + o_s + bn2 + bs2) ----
#pragma unroll
    for (int nt = 0; nt < 4; ++nt) {
        const int nn = nt * 16;
        const float bias  = bn2[nn + col] + bs2[nn + col];
        const int   mbase = (lane < 16) ? 0 : 8;
#pragma unroll
        for (int j = 0; j < 8; ++j) {
            const int i = rbase + mbase + j;
            if (i < T) out[(size_t)i * PC_D + nn + col] = fmaxf(acc2[nt][j] + bias, 0.f);
        }
    }
}

// ---------------------------------------------------------------------------
// Host-side launch
// ---------------------------------------------------------------------------
static inline size_t align256(size_t x) { return (x + 255) & ~(size_t)255; }

extern "C" void kernel_launch(void* const* d_in, const int* in_sizes, int n_in,
                              void* d_out, int out_size, void* d_ws, size_t ws_size,
                              hipStream_t stream) {
    const float* h    = (const float*)d_in[0];
    const float* cell = (const float*)d_in[1];
    const float* attn = (const float*)d_in[2];
    const float* Wn1  = (const float*)d_in[3];
    const float* bn1  = (const float*)d_in[4];
    const float* Wn2  = (const float*)d_in[5];
    const float* bn2  = (const float*)d_in[6];
    const float* Ws1  = (const float*)d_in[7];
    const float* bs1  = (const float*)d_in[8];
    const float* Ws2  = (const float*)d_in[9];
    const float* bs2  = (const float*)d_in[10];
    const int*   src  = (const int*)d_in[11];
    const int*   dst  = (const int*)d_in[12];
    const int*   tgt  = (const int*)d_in[13];
    float*       out  = (float*)d_out;

    const int N = in_sizes[0] / PC_D;
    const int E = in_sizes[11];
    const int T = in_sizes[13];

    // Workspace carve-up (~40 MB total).
    char* ws = (char*)d_ws;
    size_t o = 0;
    float*    s_sc  = (float*)(ws + o);  o = align256(o + (size_t)N * 4);
    float*    m_c   = (float*)(ws + o);  o = align256(o + (size_t)T * 4);
    float*    den_c = (float*)(ws + o);  o = align256(o + (size_t)T * 4);
    int*      mark  = (int*)(ws + o);    o = align256(o + (size_t)N * 4);
    int*      map   = (int*)(ws + o);    o = align256(o + (size_t)N * 4);
    unsigned* ctr   = (unsigned*)(ws + o); o = align256(o + 256);
    float*    ac    = (float*)(ws + o);  o = align256(o + (size_t)E * 4);
    float*    hn    = (float*)(ws + o);  o = align256(o + (size_t)T * PC_D * 4);

    const int B = 256;
    const unsigned gN = (unsigned)((N + B - 1) / B);
    const unsigned gE = (unsigned)((E + B - 1) / B);
    const unsigned gT = (unsigned)((T + B - 1) / B);
    const long long scat = (long long)E * 16;
    const unsigned gS = (unsigned)((scat + B - 1) / B);

    k_init   <<<gN, B, 0, stream>>>(mark, ctr, N);
    k_mark   <<<gT, B, 0, stream>>>(tgt, mark, T);
    k_assign <<<gN, B, 0, stream>>>(mark, map, ctr, m_c, den_c, hn, N);
    k_score  <<<gN, B, 0, stream>>>(h, attn, s_sc, N);
    k_max    <<<gE, B, 0, stream>>>(src, dst, map, s_sc, m_c, E);
    k_denom  <<<gE, B, 0, stream>>>(src, dst, map, s_sc, m_c, den_c, E);
    k_coef   <<<gE, B, 0, stream>>>(src, dst, map, s_sc, m_c, den_c, ac, E);
    k_scatter<<<gS, B, 0, stream>>>(src, dst, map, ac, h, hn, E);

    const unsigned gM = (unsigned)((T + 63) / 64);
    k_mlp<<<gM, 128, 0, stream>>>(hn, cell, Wn1, bn1, Wn2, bn2, Ws1, bs1, Ws2, bs2,
                                  tgt, map, out, T);
}